// MaxEntMeanField_42417097015741
// MI455X (gfx1250) — compile-verified
//
#include <hip/hip_runtime.h>
#include <math.h>

#define NMAT  4096
#define NN    ((size_t)NMAT * (size_t)NMAT)
#define MHIST 5
#define MAXIT 200
#define BLK   256

typedef __attribute__((ext_vector_type(2))) float v2f;
typedef __attribute__((ext_vector_type(8))) float v8f;
typedef __attribute__((ext_vector_type(4))) int   v4i;

// ---------------------------------------------------------------------------
// Async global->LDS staging (GLOBAL_LOAD_ASYNC_TO_LDS_B128, ASYNCcnt-tracked).
// Probe round 2: first param is AS1 pointer to int4 (16B vector); use matching
// address-space-qualified vector pointer types for both args.
// ---------------------------------------------------------------------------
typedef __attribute__((address_space(1))) v4i* as1_v4i_p;
typedef __attribute__((address_space(3))) v4i* as3_v4i_p;

#if __has_builtin(__builtin_amdgcn_global_load_async_to_lds_b128)
#define ASYNC_STAGE 1
__device__ __forceinline__ void async_cp16(float* lds, const float* g) {
  __builtin_amdgcn_global_load_async_to_lds_b128(
      (as1_v4i_p)(void*)g, (as3_v4i_p)(void*)lds, 0, 0);
}
__device__ __forceinline__ void async_wait0() {
#if __has_builtin(__builtin_amdgcn_s_wait_asynccnt)
  __builtin_amdgcn_s_wait_asynccnt(0);
#else
  asm volatile("s_wait_asynccnt 0x0" ::: "memory");
#endif
}
#else
#define ASYNC_STAGE 0
#endif

// ---------------------------------------------------------------------------
// WMMA helpers: V_WMMA_F32_16X16X4_F32  (D(16x16,f32) = A(16x4) @ B(4x16) + C)
// A layout (ISA 7.12.2): lanes 0-15 row M=lane, VGPR0=K0,VGPR1=K1;
//                        lanes 16-31 row M=lane-16, VGPR0=K2,VGPR1=K3.
// C layout: VGPR v -> M = v + 8*(lane>=16), N = lane&15.
// ---------------------------------------------------------------------------
__device__ __forceinline__ v8f wmma4(v2f a, v2f b, v8f c) {
  return __builtin_amdgcn_wmma_f32_16x16x4_f32(
      /*neg_a=*/false, a, /*neg_b=*/false, b,
      /*c_mod=*/(short)0, c, /*reuse_a=*/false, /*reuse_b=*/false);
}

__device__ __forceinline__ v2f ldA(const float* s, int stride, int row0, int k0, int lane) {
  int r  = row0 + (lane & 15);
  int kb = k0 + ((lane >> 4) << 1);
  v2f a;
  a.x = s[r * stride + kb];
  a.y = s[r * stride + kb + 1];
  return a;
}

__device__ __forceinline__ v2f ldB(const float* s, int stride, int k0, int col0, int lane) {
  int c  = col0 + (lane & 15);
  int kb = k0 + ((lane >> 4) << 1);
  v2f b;
  b.x = s[kb * stride + c];
  b.y = s[(kb + 1) * stride + c];
  return b;
}

// ---------------------------------------------------------------------------
// Setup: Js = 0.5*(J+J^T) with zero diagonal; J2 = Js*Js (elementwise)
// ---------------------------------------------------------------------------
__global__ void sym_kernel(const float* __restrict__ J, float* __restrict__ Js,
                           float* __restrict__ J2) {
  int idx = blockIdx.x * BLK + threadIdx.x;
  int i = idx >> 12, j = idx & (NMAT - 1);
  float v = (i == j) ? 0.0f : 0.5f * (J[idx] + J[j * NMAT + i]);
  Js[idx] = v;
  J2[idx] = v * v;
}

__global__ void init_state(const float* __restrict__ h, float* __restrict__ x,
                           float* __restrict__ xnl, int* __restrict__ flag) {
  int i = blockIdx.x * BLK + threadIdx.x;
  float v = tanhf(h[i]);
  x[i] = v;
  xnl[i] = v;
  if (i == 0) *flag = 0;
}

// ---------------------------------------------------------------------------
// TAP loop kernels (bandwidth-bound; both matrices are L2-resident: 134MB<192MB)
// ---------------------------------------------------------------------------
__global__ void dual_matvec(const float* __restrict__ Js, const float* __restrict__ J2,
                            const float* __restrict__ x, const float* __restrict__ h,
                            float* __restrict__ field, float* __restrict__ reac,
                            const int* __restrict__ flag) {
  if (*flag) return;
  int r = blockIdx.x, tid = threadIdx.x;
  const float* jr  = Js + (size_t)r * NMAT;
  const float* j2r = J2 + (size_t)r * NMAT;
  float s1 = 0.f, s2 = 0.f;
  for (int c = tid * 4; c < NMAT; c += BLK * 4) {
    __builtin_prefetch(jr + c + BLK * 4, 0, 1);
    __builtin_prefetch(j2r + c + BLK * 4, 0, 1);
    float4 a  = *(const float4*)(jr + c);
    float4 b  = *(const float4*)(j2r + c);
    float4 xv = *(const float4*)(x + c);
    s1 += a.x * xv.x + a.y * xv.y + a.z * xv.z + a.w * xv.w;
    s2 += b.x * (1.f - xv.x * xv.x) + b.y * (1.f - xv.y * xv.y) +
          b.z * (1.f - xv.z * xv.z) + b.w * (1.f - xv.w * xv.w);
  }
  __shared__ float r1[BLK];
  __shared__ float r2[BLK];
  r1[tid] = s1; r2[tid] = s2;
  __syncthreads();
  for (int s = BLK / 2; s > 0; s >>= 1) {
    if (tid < s) { r1[tid] += r1[tid + s]; r2[tid] += r2[tid + s]; }
    __syncthreads();
  }
  if (tid == 0) { field[r] = h[r] + r1[0]; reac[r] = r2[0]; }
}

__global__ void tap_residual(const float* __restrict__ field, const float* __restrict__ reac,
                             const float* __restrict__ x, float* __restrict__ xnew,
                             float* __restrict__ res, float* __restrict__ Fmat,
                             float* __restrict__ bpart, const int* __restrict__ flag, int it) {
  if (*flag) return;
  int i = blockIdx.x * BLK + threadIdx.x;
  float xi = x[i];
  float xn = tanhf(field[i] - xi * reac[i]);
  float r  = xn - xi;
  xnew[i] = xn;
  res[i]  = r;
  Fmat[i * MHIST + (it % MHIST)] = r;
  __shared__ float red[BLK];
  red[threadIdx.x] = r * r;
  __syncthreads();
  for (int s = BLK / 2; s > 0; s >>= 1) {
    if (threadIdx.x < s) red[threadIdx.x] += red[threadIdx.x + s];
    __syncthreads();
  }
  if (threadIdx.x == 0) bpart[blockIdx.x] = red[0];
}

// One workgroup: FtF (5x5) and rhs (5) via block reduction, 5x5 solve on lane 0
__global__ void anderson_kernel(const float* __restrict__ Fmat, const float* __restrict__ res,
                                float* __restrict__ alpha, const int* __restrict__ flag, int it) {
  if (*flag) return;
  int tid = threadIdx.x;
  int hist = min(it + 1, MHIST);
  float acc[30];
#pragma unroll
  for (int v = 0; v < 30; ++v) acc[v] = 0.f;
  for (int i = tid; i < NMAT; i += BLK) {
    float f[MHIST];
#pragma unroll
    for (int c = 0; c < MHIST; ++c) f[c] = (c < hist) ? Fmat[i * MHIST + c] : 0.f;
    float ri = res[i];
#pragma unroll
    for (int a = 0; a < MHIST; ++a) {
#pragma unroll
      for (int b = 0; b < MHIST; ++b) acc[a * MHIST + b] += f[a] * f[b];
      acc[25 + a] += f[a] * ri;
    }
  }
  __shared__ float red[BLK];
  __shared__ float tot[30];
  for (int v = 0; v < 30; ++v) {
    red[tid] = acc[v];
    __syncthreads();
    for (int s = BLK / 2; s > 0; s >>= 1) {
      if (tid < s) red[tid] += red[tid + s];
      __syncthreads();
    }
    if (tid == 0) tot[v] = red[0];
    __syncthreads();
  }
  if (tid == 0) {
    float M[MHIST][MHIST], b[MHIST];
    for (int a = 0; a < MHIST; ++a) {
      for (int c = 0; c < MHIST; ++c) M[a][c] = tot[a * MHIST + c];
      M[a][a] += 1e-6f + ((a < hist) ? 0.f : 1.f);
      b[a] = tot[25 + a];
    }
    for (int k = 0; k < MHIST; ++k) {
      float ip = 1.0f / M[k][k];
      for (int c = k; c < MHIST; ++c) M[k][c] *= ip;
      b[k] *= ip;
      for (int r2 = 0; r2 < MHIST; ++r2) {
        if (r2 == k) continue;
        float fm = M[r2][k];
        for (int c = k; c < MHIST; ++c) M[r2][c] -= fm * M[k][c];
        b[r2] -= fm * b[k];
      }
    }
    for (int a = 0; a < MHIST; ++a) alpha[a] = b[a];
  }
}

__global__ void mix_kernel(float* __restrict__ x, const float* __restrict__ xnew,
                           float* __restrict__ xnl, const float* __restrict__ Fmat,
                           const float* __restrict__ alpha, const int* __restrict__ flag, int it) {
  if (*flag) return;
  int i = blockIdx.x * BLK + threadIdx.x;
  int hist = min(it + 1, MHIST);
  float ex = 0.f;
#pragma unroll
  for (int c = 0; c < MHIST; ++c)
    if (c < hist) ex += Fmat[i * MHIST + c] * alpha[c];
  float xi = x[i];
  float xa = xi + 0.5f * ex;                 // DAMPING*x + (1-D)*(x + BETA*extrap)
  float xp = 0.5f * xi + 0.5f * xnew[i];     // DAMPING*x + (1-D)*x_new
  x[i] = (hist >= 2) ? xa : xp;
  xnl[i] = xnew[i];
}

__global__ void finalize_kernel(const float* __restrict__ bpart, int* __restrict__ flag) {
  if (threadIdx.x == 0 && blockIdx.x == 0 && !*flag) {
    float s = 0.f;
    for (int b = 0; b < NMAT / BLK; ++b) s += bpart[b];
    if (sqrtf(s) / (float)NMAT < 1e-5f) *flag = 1;
  }
}

__global__ void select_m(const float* __restrict__ x, const float* __restrict__ xnl,
                         const int* __restrict__ flag, float* __restrict__ m,
                         float* __restrict__ out) {
  int i = blockIdx.x * BLK + threadIdx.x;
  float v = (*flag) ? xnl[i] : x[i];
  m[i] = v;
  out[i] = v;
}

__global__ void build_A(const float* __restrict__ Js, const float* __restrict__ m,
                        float* __restrict__ A) {
  int idx = blockIdx.x * BLK + threadIdx.x;
  int i = idx >> 12, j = idx & (NMAT - 1);
  float v = -Js[idx];
  if (i == j) {
    float mi = m[i];
    v += 1.0f / (1.0f - mi * mi + 1e-8f) + 1e-6f;
  }
  A[idx] = v;
}

// ---------------------------------------------------------------------------
// Blocked in-place Gauss-Jordan inversion, NB = 64
// ---------------------------------------------------------------------------
__global__ void copy_panel(const float* __restrict__ A, float* __restrict__ panel, int kb) {
  int idx = blockIdx.x * BLK + threadIdx.x;   // N*64 entries
  int i = idx >> 6, c = idx & 63;
  panel[idx] = A[i * NMAT + kb * 64 + c];
}

// Invert the 64x64 diagonal block: Gauss-Jordan on [M | I] in LDS
__global__ void invert_diag(const float* __restrict__ A, float* __restrict__ P, int kb) {
  __shared__ float aug[64 * 130];
  __shared__ float colk[64];
  int tid = threadIdx.x;
  for (int idx = tid; idx < 64 * 64; idx += BLK) {
    int r = idx >> 6, c = idx & 63;
    aug[r * 130 + c]      = A[(kb * 64 + r) * NMAT + kb * 64 + c];
    aug[r * 130 + 64 + c] = (r == c) ? 1.0f : 0.0f;
  }
  __syncthreads();
  for (int k = 0; k < 64; ++k) {
    float ip = 1.0f / aug[k * 130 + k];
    __syncthreads();
    if (tid < 128) aug[k * 130 + tid] *= ip;
    __syncthreads();
    if (tid < 64) colk[tid] = aug[tid * 130 + k];
    __syncthreads();
    for (int idx = tid; idx < 64 * 128; idx += BLK) {
      int r = idx >> 7, c = idx & 127;
      if (r != k) aug[r * 130 + c] -= colk[r] * aug[k * 130 + c];
    }
    __syncthreads();
  }
  for (int idx = tid; idx < 64 * 64; idx += BLK) {
    int r = idx >> 6, c = idx & 63;
    P[idx] = aug[r * 130 + 64 + c];
  }
}

// A[kb rows, colTile] = P @ A[kb rows, colTile]   (64x64 tiles, WMMA)
__global__ void gj_rowupdate(float* __restrict__ A, const float* __restrict__ P, int kb) {
  int colTile = blockIdx.x;
  __shared__ __align__(16) float lP[64 * 68];
  __shared__ __align__(16) float lB[64 * 68];
  int tid = threadIdx.x;
#if ASYNC_STAGE
  for (int idx = tid; idx < 64 * 16; idx += BLK) {         // 16B chunks
    int r = idx >> 4, c4 = (idx & 15) << 2;
    async_cp16(&lP[r * 68 + c4], &P[r * 64 + c4]);
    async_cp16(&lB[r * 68 + c4], &A[(kb * 64 + r) * NMAT + colTile * 64 + c4]);
  }
  async_wait0();
#else
  for (int idx = tid; idx < 64 * 64; idx += BLK) {
    int r = idx >> 6, c = idx & 63;
    lP[r * 68 + c] = P[idx];
    lB[r * 68 + c] = A[(kb * 64 + r) * NMAT + colTile * 64 + c];
  }
#endif
  __syncthreads();
  int wave = tid >> 5, lane = tid & 31;
  int wr = (wave & 3) * 16;        // 4 wave-rows  x 16 = 64
  int wc = (wave >> 2) * 32;       // 2 wave-cols x 32 = 64
  v8f acc0 = {}, acc1 = {};
#pragma unroll
  for (int k0 = 0; k0 < 64; k0 += 4) {
    v2f a  = ldA(lP, 68, wr, k0, lane);
    v2f b0 = ldB(lB, 68, k0, wc, lane);
    v2f b1 = ldB(lB, 68, k0, wc + 16, lane);
    acc0 = wmma4(a, b0, acc0);
    acc1 = wmma4(a, b1, acc1);
  }
  int rbase = kb * 64 + wr + ((lane >> 4) << 3);
  int c0 = colTile * 64 + wc + (lane & 15);
#pragma unroll
  for (int v = 0; v < 8; ++v) {
    A[(rbase + v) * NMAT + c0]      = acc0[v];
    A[(rbase + v) * NMAT + c0 + 16] = acc1[v];
  }
}

// Trailing/leading update: A[rowTile, colTile] -= panel[rowTile] @ A[kb, colTile]
// Workgroup tile = 64 rows x 128 cols; 8 waves x (32x32 each, 2x2 WMMA accumulators)
__global__ void gj_update(float* __restrict__ A, const float* __restrict__ panel, int kb) {
  int rowTile = blockIdx.y;
  if (rowTile == kb) return;
  int colTile = blockIdx.x;
  __shared__ __align__(16) float lA[64 * 68];
  __shared__ __align__(16) float lB[64 * 132];
  int tid = threadIdx.x;
#if ASYNC_STAGE
  for (int idx = tid; idx < 64 * 16; idx += BLK) {         // A: 16B chunks
    int r = idx >> 4, c4 = (idx & 15) << 2;
    async_cp16(&lA[r * 68 + c4], &panel[(rowTile * 64 + r) * 64 + c4]);
  }
  for (int idx = tid; idx < 64 * 32; idx += BLK) {         // B: 16B chunks
    int r = idx >> 5, c4 = (idx & 31) << 2;
    async_cp16(&lB[r * 132 + c4], &A[(kb * 64 + r) * NMAT + colTile * 128 + c4]);
  }
  async_wait0();
#else
  for (int idx = tid; idx < 64 * 64; idx += BLK) {
    int r = idx >> 6, c = idx & 63;
    lA[r * 68 + c] = panel[(rowTile * 64 + r) * 64 + c];
  }
  for (int idx = tid; idx < 64 * 128; idx += BLK) {
    int r = idx >> 7, c = idx & 127;
    lB[r * 132 + c] = A[(kb * 64 + r) * NMAT + colTile * 128 + c];
  }
#endif
  __syncthreads();
  int wave = tid >> 5, lane = tid & 31;
  int wr = (wave & 1) * 32;        // 2 wave-rows x 32 = 64
  int wc = (wave >> 1) * 32;       // 4 wave-cols x 32 = 128
  v8f acc00 = {}, acc01 = {}, acc10 = {}, acc11 = {};
#pragma unroll
  for (int k0 = 0; k0 < 64; k0 += 4) {
    v2f a0 = ldA(lA, 68, wr, k0, lane);
    v2f a1 = ldA(lA, 68, wr + 16, k0, lane);
    v2f b0 = ldB(lB, 132, k0, wc, lane);
    v2f b1 = ldB(lB, 132, k0, wc + 16, lane);
    acc00 = wmma4(a0, b0, acc00);
    acc01 = wmma4(a0, b1, acc01);
    acc10 = wmma4(a1, b0, acc10);
    acc11 = wmma4(a1, b1, acc11);
  }
  int col = lane & 15;
  int rsel = (lane >> 4) << 3;
  int gr0 = rowTile * 64 + wr + rsel;
  int gr1 = rowTile * 64 + wr + 16 + rsel;
  int gc0 = colTile * 128 + wc + col;
#pragma unroll
  for (int v = 0; v < 8; ++v) {
    A[(gr0 + v) * NMAT + gc0]      -= acc00[v];
    A[(gr0 + v) * NMAT + gc0 + 16] -= acc01[v];
    A[(gr1 + v) * NMAT + gc0]      -= acc10[v];
    A[(gr1 + v) * NMAT + gc0 + 16] -= acc11[v];
  }
}

// Column write: A[rowTile, kb] = -panel[rowTile] @ P  (and A[kb][kb] = P)
__global__ void gj_colwrite(float* __restrict__ A, const float* __restrict__ panel,
                            const float* __restrict__ P, int kb) {
  int rowTile = blockIdx.x;
  int tid = threadIdx.x;
  if (rowTile == kb) {
    for (int idx = tid; idx < 64 * 64; idx += BLK) {
      int r = idx >> 6, c = idx & 63;
      A[(kb * 64 + r) * NMAT + kb * 64 + c] = P[idx];
    }
    return;
  }
  __shared__ __align__(16) float lA[64 * 68];
  __shared__ __align__(16) float lP[64 * 68];
#if ASYNC_STAGE
  for (int idx = tid; idx < 64 * 16; idx += BLK) {
    int r = idx >> 4, c4 = (idx & 15) << 2;
    async_cp16(&lA[r * 68 + c4], &panel[(rowTile * 64 + r) * 64 + c4]);
    async_cp16(&lP[r * 68 + c4], &P[r * 64 + c4]);
  }
  async_wait0();
#else
  for (int idx = tid; idx < 64 * 64; idx += BLK) {
    int r = idx >> 6, c = idx & 63;
    lA[r * 68 + c] = panel[(rowTile * 64 + r) * 64 + c];
    lP[r * 68 + c] = P[idx];
  }
#endif
  __syncthreads();
  int wave = tid >> 5, lane = tid & 31;
  int wr = (wave & 3) * 16;
  int wc = (wave >> 2) * 32;
  v8f acc0 = {}, acc1 = {};
#pragma unroll
  for (int k0 = 0; k0 < 64; k0 += 4) {
    v2f a  = ldA(lA, 68, wr, k0, lane);
    v2f b0 = ldB(lP, 68, k0, wc, lane);
    v2f b1 = ldB(lP, 68, k0, wc + 16, lane);
    acc0 = wmma4(a, b0, acc0);
    acc1 = wmma4(a, b1, acc1);
  }
  int rbase = rowTile * 64 + wr + ((lane >> 4) << 3);
  int c0 = kb * 64 + wc + (lane & 15);
#pragma unroll
  for (int v = 0; v < 8; ++v) {
    A[(rbase + v) * NMAT + c0]      = -acc0[v];
    A[(rbase + v) * NMAT + c0 + 16] = -acc1[v];
  }
}

// cov upper triangle: out[N + triu(i,j)] = m_i*m_j + chi[i][j]
__global__ void cov_out(const float* __restrict__ m, const float* __restrict__ chi,
                        float* __restrict__ out) {
  int idx = blockIdx.x * BLK + threadIdx.x;
  int i = idx >> 12, j = idx & (NMAT - 1);
  if (j <= i) return;
  int t = i * (NMAT - 1) - (i * (i - 1)) / 2 + (j - i - 1);
  out[NMAT + t] = m[i] * m[j] + chi[idx];
}

// ---------------------------------------------------------------------------
extern "C" void kernel_launch(void* const* d_in, const int* in_sizes, int n_in,
                              void* d_out, int out_size, void* d_ws, size_t ws_size,
                              hipStream_t stream) {
  (void)in_sizes; (void)n_in; (void)out_size; (void)ws_size;
  const float* h = (const float*)d_in[0];
  const float* J = (const float*)d_in[1];
  float* out = (float*)d_out;

  float* ws    = (float*)d_ws;
  float* Js    = ws;                          // N*N
  float* A     = ws + NN;                     // N*N  (J2 during loop, then A, then chi)
  float* panel = ws + 2 * NN;                 // N*64
  float* P     = panel + (size_t)NMAT * 64;   // 64*64
  float* field = P + 64 * 64;
  float* reac  = field + NMAT;
  float* x     = reac + NMAT;
  float* xnew  = x + NMAT;
  float* xnl   = xnew + NMAT;
  float* res   = xnl + NMAT;
  float* Fmat  = res + NMAT;                  // N*5
  float* alpha = Fmat + (size_t)NMAT * MHIST; // 8
  float* bpart = alpha + 8;                   // 16
  float* mvec  = bpart + 16;                  // N
  int*   flag  = (int*)(mvec + NMAT);

  const int nvBlocks = NMAT / BLK;            // 16
  const int nnBlocks = (int)(NN / BLK);       // 65536

  sym_kernel<<<nnBlocks, BLK, 0, stream>>>(J, Js, A);
  init_state<<<nvBlocks, BLK, 0, stream>>>(h, x, xnl, flag);

  for (int it = 0; it < MAXIT; ++it) {
    dual_matvec<<<NMAT, BLK, 0, stream>>>(Js, A, x, h, field, reac, flag);
    tap_residual<<<nvBlocks, BLK, 0, stream>>>(field, reac, x, xnew, res, Fmat, bpart, flag, it);
    anderson_kernel<<<1, BLK, 0, stream>>>(Fmat, res, alpha, flag, it);
    mix_kernel<<<nvBlocks, BLK, 0, stream>>>(x, xnew, xnl, Fmat, alpha, flag, it);
    finalize_kernel<<<1, 64, 0, stream>>>(bpart, flag);
  }

  select_m<<<nvBlocks, BLK, 0, stream>>>(x, xnl, flag, mvec, out);
  build_A<<<nnBlocks, BLK, 0, stream>>>(Js, mvec, A);

  for (int kb = 0; kb < NMAT / 64; ++kb) {
    copy_panel<<<NMAT * 64 / BLK, BLK, 0, stream>>>(A, panel, kb);
    invert_diag<<<1, BLK, 0, stream>>>(A, P, kb);
    gj_rowupdate<<<NMAT / 64, BLK, 0, stream>>>(A, P, kb);
    gj_update<<<dim3(NMAT / 128, NMAT / 64), BLK, 0, stream>>>(A, panel, kb);
    gj_colwrite<<<NMAT / 64, BLK, 0, stream>>>(A, panel, P, kb);
  }

  cov_out<<<nnBlocks, BLK, 0, stream>>>(mvec, A, out);
}